// F_nonlocal_72335839200045
// MI455X (gfx1250) — compile-verified
//
#include <hip/hip_runtime.h>
#include <hip/hip_bf16.h>
#include <math.h>

// ---------------- problem constants (from reference) ----------------
constexpr int Bq = 4, Nq = 512, Pq = 1024, Kq = 20;
constexpr int Eq  = Nq * Kq;        // 10240 atom edges / batch
constexpr int PEq = Pq * Kq;        // 20480 probe edges / batch
constexpr int NA  = Bq * Nq;        // 2048 atom nodes
constexpr int NP  = Bq * Pq;        // 4096 probe nodes
constexpr int ET  = Bq * Eq;        // 40960 atom edges
constexpr int PET = Bq * PEq;       // 81920 probe edges
constexpr int MUL = 64, SHD = 9, NBq = 20, HID = 100, Tq = 3;
constexpr int FEAT = SHD * MUL;     // 576 floats per node
constexpr int EW   = 4;             // waves per edge-MLP block

typedef float v2f __attribute__((ext_vector_type(2)));
typedef float v8f __attribute__((ext_vector_type(8)));
typedef unsigned int u32x4 __attribute__((ext_vector_type(4)));
typedef int i32x4 __attribute__((ext_vector_type(4)));
typedef int i32x8 __attribute__((ext_vector_type(8)));

// D(16x16 f32) += A(16x4 f32) * B(4x16 f32)  — full-precision matrix pipe
__device__ __forceinline__ v8f wmma4_f32(v2f a, v2f b, v8f c) {
  return __builtin_amdgcn_wmma_f32_16x16x4_f32(
      /*neg_a=*/false, a, /*neg_b=*/false, b,
      /*c_mod=*/(short)0, c, /*reuse_a=*/false, /*reuse_b=*/false);
}

__device__ __forceinline__ float silu_f(float v) { return v / (1.0f + __expf(-v)); }
__device__ __forceinline__ float sigm_f(float v) { return 1.0f / (1.0f + __expf(-v)); }

// raw LDS byte offset of a __shared__ object
__device__ __forceinline__ unsigned int lds_offset(const void* p) {
  return (unsigned int)(unsigned long long)(__attribute__((address_space(3))) const char*)p;
}

// TDM: DMA nelem f32 from global -> LDS (1-row tile), tracked by TENSORcnt.
// D# layout per CDNA5 ISA.
__device__ __forceinline__ void tdm_load_f32(const void* gptr, unsigned int lds_addr,
                                             unsigned int nelem) {
  unsigned long long ga = (unsigned long long)gptr;
  u32x4 g0;
  g0[0] = 1u;                                            // count=1, user mode
  g0[1] = lds_addr;                                      // lds_addr
  g0[2] = (unsigned int)ga;                              // global_addr[31:0]
  g0[3] = (unsigned int)((ga >> 32) & 0x1FFFFFFu) | (2u << 30);  // addr[56:32] | type=2
  i32x8 g1;
  g1[0] = (int)(2u << 16);                               // data_size=4B, no multicast
  g1[1] = (int)((nelem & 0xFFFFu) << 16);                // tensor_dim0[15:0]
  g1[2] = (int)(((nelem >> 16) & 0xFFFFu) | (1u << 16)); // dim0[31:16] | tensor_dim1=1
  g1[3] = (int)((nelem & 0xFFFFu) << 16);                // tile_dim0 = nelem
  g1[4] = 1;                                             // tile_dim1=1, tile_dim2=0
  g1[5] = (int)nelem;                                    // tensor_dim0_stride
  g1[6] = 0;
  g1[7] = 0;
  i32x4 z4 = {0, 0, 0, 0};
#if __clang_major__ >= 23
  i32x8 z8 = {0, 0, 0, 0, 0, 0, 0, 0};
  __builtin_amdgcn_tensor_load_to_lds(g0, g1, z4, z4, z8, 0);
#else
  __builtin_amdgcn_tensor_load_to_lds(g0, g1, z4, z4, 0);
#endif
}

// ---------------- geometry: edge vec -> sh[9], rbf[20] ----------------
__global__ void geom_kernel(const float* __restrict__ pos_src,
                            const float* __restrict__ pos_dst,
                            const int*   __restrict__ edges,   // (B,EPB,2)
                            const float* __restrict__ disp,    // (B,EPB,3)
                            const float* __restrict__ cell,    // (B,3,3)
                            int epb, int nTot, int srcPerB, int dstPerB,
                            int* __restrict__ gsrc, int* __restrict__ gdst,
                            float* __restrict__ sh, float* __restrict__ rbf) {
  int e = blockIdx.x * blockDim.x + threadIdx.x;
  if (e >= nTot) return;
  int b = e / epb;
  int gs = b * srcPerB + edges[2 * e + 0];
  int gd = b * dstPerB + edges[2 * e + 1];
  gsrc[e] = gs; gdst[e] = gd;
  const float* C = cell + b * 9;
  float d0 = disp[3 * e], d1 = disp[3 * e + 1], d2 = disp[3 * e + 2];
  float dx = d0 * C[0] + d1 * C[3] + d2 * C[6];
  float dy = d0 * C[1] + d1 * C[4] + d2 * C[7];
  float dz = d0 * C[2] + d1 * C[5] + d2 * C[8];
  float vx = pos_dst[3 * gd + 0] - (pos_src[3 * gs + 0] + dx);
  float vy = pos_dst[3 * gd + 1] - (pos_src[3 * gs + 1] + dy);
  float vz = pos_dst[3 * gd + 2] - (pos_src[3 * gs + 2] + dz);
  float r = sqrtf(vx * vx + vy * vy + vz * vz);
  float inv = 1.0f / fmaxf(r, 1e-9f);
  float x = vx * inv, y = vy * inv, z = vz * inv;
  const float s3 = 1.7320508075688772f, s5 = 2.23606797749979f, s15 = 3.872983346207417f;
  float* S = sh + (size_t)e * 9;
  S[0] = 1.0f;
  S[1] = s3 * x; S[2] = s3 * y; S[3] = s3 * z;
  S[4] = s15 * x * y; S[5] = s15 * y * z;
  S[6] = 0.5f * s5 * (3.0f * z * z - 1.0f);
  S[7] = s15 * x * z;
  S[8] = 0.5f * s15 * (x * x - y * y);
  float* R = rbf + (size_t)e * 20;
#pragma unroll
  for (int i = 0; i < 20; ++i) {
    float t = (r - 4.0f * (float)i / 19.0f) * 5.0f;  // sigma = 0.2
    R[i] = __expf(-t * t);
  }
}

// ---------------- h init: one-hot embedding ----------------
__global__ void embed_kernel(const float* __restrict__ W_embed,
                             const int* __restrict__ species,
                             float* __restrict__ h) {
  int tid = blockIdx.x * blockDim.x + threadIdx.x;
  if (tid >= NA * FEAT) return;
  int n = tid / FEAT, r = tid % FEAT;
  h[tid] = (r < MUL) ? W_embed[species[n] * MUL + r] : 0.0f;
}

// ------- edge MLP: w = silu(rbf@Wr1+br1)@Wr2+br2  (WMMA f32, TDM-staged) ----
__global__ __launch_bounds__(EW * 32)
void edge_mlp_kernel(const float* __restrict__ rbf,
                     const float* __restrict__ Wr1, const float* __restrict__ br1,
                     const float* __restrict__ Wr2, const float* __restrict__ br2,
                     float* __restrict__ wout) {
  __shared__ float lWr1[NBq * HID];        // 20x100
  __shared__ float lbr1[HID];
  __shared__ float lWr2[HID * MUL];        // 100x64
  __shared__ float lbr2[MUL];
  __shared__ float hid[EW][16 * 112];      // per-wave hidden tile

  const int wid  = threadIdx.x >> 5;
  const int lane = threadIdx.x & 31;
  const int lm   = lane & 15;
  const int half = lane >> 4;
  const int eBase = blockIdx.x * (EW * 16) + wid * 16;
  const size_t arow = (size_t)(eBase + lm) * 20;

  // wave 0 DMAs all layer weights into LDS via the Tensor Data Mover
  if (wid == 0) {
    tdm_load_f32(Wr1, lds_offset(lWr1), NBq * HID);
    tdm_load_f32(br1, lds_offset(lbr1), HID);
    tdm_load_f32(Wr2, lds_offset(lWr2), HID * MUL);
    tdm_load_f32(br2, lds_offset(lbr2), MUL);
    __builtin_amdgcn_s_wait_tensorcnt(0);
  }
  __syncthreads();

  // GEMM1: [16x20] @ [20x100] (+bias, silu) -> LDS [16x112]
  for (int nt = 0; nt < 7; ++nt) {
    int n = nt * 16 + lm;
    bool nv = (n < HID);
    float bias = nv ? lbr1[nv ? n : 0] : 0.0f;
    v8f acc;
#pragma unroll
    for (int i = 0; i < 8; ++i) acc[i] = bias;
#pragma unroll
    for (int kc = 0; kc < 5; ++kc) {
      int k0 = kc * 4 + half * 2;
      v2f a, bv;
      a.x = rbf[arow + k0];
      a.y = rbf[arow + k0 + 1];
      bv.x = nv ? lWr1[k0 * HID + (nv ? n : 0)] : 0.0f;
      bv.y = nv ? lWr1[(k0 + 1) * HID + (nv ? n : 0)] : 0.0f;
      acc = wmma4_f32(a, bv, acc);
    }
#pragma unroll
    for (int i = 0; i < 8; ++i) {
      int m = half * 8 + i;
      hid[wid][m * 112 + nt * 16 + lm] = silu_f(acc[i]);
    }
  }
  __syncthreads();

  // GEMM2: [16x100] @ [100x64] (+bias) -> w
  for (int nt = 0; nt < 4; ++nt) {
    int n = nt * 16 + lm;
    float bias = lbr2[n];
    v8f acc;
#pragma unroll
    for (int i = 0; i < 8; ++i) acc[i] = bias;
#pragma unroll
    for (int kc = 0; kc < 25; ++kc) {
      int k0 = kc * 4 + half * 2;
      v2f a, bv;
      a.x = hid[wid][lm * 112 + k0];
      a.y = hid[wid][lm * 112 + k0 + 1];
      bv.x = lWr2[k0 * MUL + n];
      bv.y = lWr2[(k0 + 1) * MUL + n];
      acc = wmma4_f32(a, bv, acc);
    }
#pragma unroll
    for (int i = 0; i < 8; ++i)
      wout[(size_t)(eBase + half * 8 + i) * MUL + n] = acc[i];
  }
}

// ---------------- scatter: a[dst] += sh ⊗ (feat[src]*w)/sqrt(K) --------
__global__ void scatter_kernel(const int* __restrict__ gsrc,
                               const int* __restrict__ gdst,
                               const float* __restrict__ sh,
                               const float* __restrict__ w,
                               const float* __restrict__ feat,
                               float* __restrict__ abuf, int nEdges) {
  int tid = blockIdx.x * blockDim.x + threadIdx.x;
  if (tid >= nEdges * MUL) return;
  int e = tid >> 6, c = tid & 63;
  float base = feat[(size_t)gsrc[e] * FEAT + c] * w[tid] * 0.22360679774997896f;
  float* dp = abuf + (size_t)gdst[e] * FEAT + c;
  const float* S = sh + (size_t)e * 9;
#pragma unroll
  for (int s = 0; s < 9; ++s)
    atomicAdd(dp + s * MUL, S[s] * base);
}

// ---------------- node update: per-l linear + gate + residual (WMMA) ----
__global__ __launch_bounds__(32)
void node_update_kernel(const float* __restrict__ abuf,
                        const float* __restrict__ Wl,  // 3 x 64 x 64
                        const float* __restrict__ Wg,  // 64 x 64
                        float* __restrict__ hio,
                        float* __restrict__ rep) {     // may be null
  const int lane = threadIdx.x;
  const int lm = lane & 15;
  const int half = lane >> 4;
  const int nb = blockIdx.x * 16;
  const size_t arow = (size_t)(nb + lm) * FEAT;

  // gate = sigmoid(a[:,0,:] @ Wg) ; 4 n-tiles of 16
  v8f gate[4];
#pragma unroll
  for (int nt = 0; nt < 4; ++nt) {
    int n = nt * 16 + lm;
    v8f acc;
#pragma unroll
    for (int i = 0; i < 8; ++i) acc[i] = 0.0f;
#pragma unroll
    for (int kc = 0; kc < 16; ++kc) {
      int k0 = kc * 4 + half * 2;
      v2f a, bv;
      a.x = abuf[arow + k0];
      a.y = abuf[arow + k0 + 1];
      bv.x = Wg[k0 * MUL + n];
      bv.y = Wg[(k0 + 1) * MUL + n];
      acc = wmma4_f32(a, bv, acc);
    }
#pragma unroll
    for (int i = 0; i < 8; ++i) gate[nt][i] = sigm_f(acc[i]);
  }

  // u[:,s,:] = a[:,s,:] @ Wl[l(s)] ; then silu (s==0) or gate, + residual
  for (int s = 0; s < 9; ++s) {
    const float* W = Wl + (size_t)((s == 0) ? 0 : (s < 4 ? 1 : 2)) * MUL * MUL;
    for (int nt = 0; nt < 4; ++nt) {
      int n = nt * 16 + lm;
      v8f acc;
#pragma unroll
      for (int i = 0; i < 8; ++i) acc[i] = 0.0f;
#pragma unroll
      for (int kc = 0; kc < 16; ++kc) {
        int k0 = kc * 4 + half * 2;
        v2f a, bv;
        a.x = abuf[arow + s * MUL + k0];
        a.y = abuf[arow + s * MUL + k0 + 1];
        bv.x = W[k0 * MUL + n];
        bv.y = W[(k0 + 1) * MUL + n];
        acc = wmma4_f32(a, bv, acc);
      }
#pragma unroll
      for (int i = 0; i < 8; ++i) {
        int m = half * 8 + i;
        float v = acc[i];
        float outv = (s == 0) ? silu_f(v) : v * gate[nt][i];
        size_t idx = (size_t)(nb + m) * FEAT + s * MUL + n;
        float nh = hio[idx] + outv;
        hio[idx] = nh;
        if (rep) rep[idx] = nh;
      }
    }
  }
}

// ---------------- readout: out = p[:,0,:] @ W_out ----------------
__global__ void out_kernel(const float* __restrict__ p,
                           const float* __restrict__ Wout,
                           float* __restrict__ out) {
  int i = blockIdx.x * blockDim.x + threadIdx.x;
  if (i >= NP) return;
  float s = 0.0f;
#pragma unroll
  for (int c = 0; c < MUL; ++c) s += p[(size_t)i * FEAT + c] * Wout[c];
  out[i] = s;
}

// ---------------- host orchestration ----------------
extern "C" void kernel_launch(void* const* d_in, const int* in_sizes, int n_in,
                              void* d_out, int out_size, void* d_ws, size_t ws_size,
                              hipStream_t stream) {
  const float* atom_xyz   = (const float*)d_in[0];
  const float* a_disp     = (const float*)d_in[1];
  const float* cell       = (const float*)d_in[2];
  const float* probe_xyz  = (const float*)d_in[3];
  const float* p_disp     = (const float*)d_in[4];
  const float* W_embed    = (const float*)d_in[5];
  const float* Wr1        = (const float*)d_in[6];
  const float* br1        = (const float*)d_in[7];
  const float* Wr2        = (const float*)d_in[8];
  const float* br2        = (const float*)d_in[9];
  const float* Wl         = (const float*)d_in[10];
  const float* Wg         = (const float*)d_in[11];
  const float* pWr1       = (const float*)d_in[12];
  const float* pbr1       = (const float*)d_in[13];
  const float* pWr2       = (const float*)d_in[14];
  const float* pbr2       = (const float*)d_in[15];
  const float* pWl        = (const float*)d_in[16];
  const float* pWg        = (const float*)d_in[17];
  const float* W_out      = (const float*)d_in[18];
  const int*   atom_edges  = (const int*)d_in[19];
  const int*   probe_edges = (const int*)d_in[20];
  const int*   species     = (const int*)d_in[21];
  float* out = (float*)d_out;

  // ---- carve workspace ----
  char* ws = (char*)d_ws;
  size_t o = 0;
  auto alloc = [&](size_t bytes) -> void* {
    void* p = ws + o;
    o += (bytes + 255) & ~(size_t)255;
    return p;
  };
  int*   gsrc_a = (int*)  alloc((size_t)ET * 4);
  int*   gdst_a = (int*)  alloc((size_t)ET * 4);
  int*   gsrc_p = (int*)  alloc((size_t)PET * 4);
  int*   gdst_p = (int*)  alloc((size_t)PET * 4);
  float* sh_a   = (float*)alloc((size_t)ET * 9 * 4);
  float* rbf_a  = (float*)alloc((size_t)ET * 20 * 4);
  float* sh_p   = (float*)alloc((size_t)PET * 9 * 4);
  float* rbf_p  = (float*)alloc((size_t)PET * 20 * 4);
  float* h      = (float*)alloc((size_t)NA * FEAT * 4);
  float* reps   = (float*)alloc((size_t)Tq * NA * FEAT * 4);
  float* abuf   = (float*)alloc((size_t)NP * FEAT * 4);   // shared atom/probe aggregation
  float* wbuf   = (float*)alloc((size_t)PET * MUL * 4);   // edge filter weights
  float* pbuf   = (float*)alloc((size_t)NP * FEAT * 4);
  (void)ws_size; (void)in_sizes; (void)n_in; (void)out_size;

  // ---- geometry ----
  geom_kernel<<<(ET + 255) / 256, 256, 0, stream>>>(
      atom_xyz, atom_xyz, atom_edges, a_disp, cell, Eq, ET, Nq, Nq,
      gsrc_a, gdst_a, sh_a, rbf_a);
  geom_kernel<<<(PET + 255) / 256, 256, 0, stream>>>(
      atom_xyz, probe_xyz, probe_edges, p_disp, cell, PEq, PET, Nq, Pq,
      gsrc_p, gdst_p, sh_p, rbf_p);

  // ---- embedding ----
  embed_kernel<<<(NA * FEAT + 255) / 256, 256, 0, stream>>>(W_embed, species, h);

  // ---- atom message-passing layers ----
  for (int t = 0; t < Tq; ++t) {
    hipMemsetAsync(abuf, 0, (size_t)NA * FEAT * 4, stream);
    edge_mlp_kernel<<<ET / (EW * 16), EW * 32, 0, stream>>>(
        rbf_a, Wr1 + (size_t)t * NBq * HID, br1 + (size_t)t * HID,
        Wr2 + (size_t)t * HID * MUL, br2 + (size_t)t * MUL, wbuf);
    scatter_kernel<<<(ET * MUL + 255) / 256, 256, 0, stream>>>(
        gsrc_a, gdst_a, sh_a, wbuf, h, abuf, ET);
    node_update_kernel<<<NA / 16, 32, 0, stream>>>(
        abuf, Wl + (size_t)t * 3 * MUL * MUL, Wg + (size_t)t * MUL * MUL,
        h, reps + (size_t)t * NA * FEAT);
  }

  // ---- probe layers ----
  hipMemsetAsync(pbuf, 0, (size_t)NP * FEAT * 4, stream);
  for (int t = 0; t < Tq; ++t) {
    hipMemsetAsync(abuf, 0, (size_t)NP * FEAT * 4, stream);
    edge_mlp_kernel<<<PET / (EW * 16), EW * 32, 0, stream>>>(
        rbf_p, pWr1 + (size_t)t * NBq * HID, pbr1 + (size_t)t * HID,
        pWr2 + (size_t)t * HID * MUL, pbr2 + (size_t)t * MUL, wbuf);
    scatter_kernel<<<(PET * MUL + 255) / 256, 256, 0, stream>>>(
        gsrc_p, gdst_p, sh_p, wbuf, reps + (size_t)t * NA * FEAT, abuf, PET);
    node_update_kernel<<<NP / 16, 32, 0, stream>>>(
        abuf, pWl + (size_t)t * 3 * MUL * MUL, pWg + (size_t)t * MUL * MUL,
        pbuf, nullptr);
  }

  // ---- readout ----
  out_kernel<<<(NP + 255) / 256, 256, 0, stream>>>(pbuf, W_out, out);
}